// LabelSmoothingLoss_46901042872882
// MI455X (gfx1250) — compile-verified
//
#include <hip/hip_runtime.h>
#include <hip/hip_bf16.h>

// ---------------------------------------------------------------------------
// Label-smoothing loss over grouped classes (B=16384, C=4432, 4 groups of 1108)
//   loss_i = lse_i - (0.9 - s)*x[t_i] - s * sum_{c in group(t_i)} x[i,c]
//   out    = mean_i loss_i,   s = 0.1/1107
// HBM-bound (~290 MB streamed once). CDNA5 path: async global->LDS staging,
// WMMA f32 16x16x4 used for the final partial-sum reduction.
// ---------------------------------------------------------------------------

#define N_CLASS   4432
#define GROUP_SZ  1108
#define BATCH     16384
#define NTHREADS  256
#define NWAVES    (NTHREADS / 32)
#define NVEC      (N_CLASS / 4)          // 1108 float4 chunks per row

static constexpr float SMOOTH_V = 0.1f / 1107.0f;
static constexpr float CONF     = 0.9f;

typedef float v2f __attribute__((ext_vector_type(2)));
typedef float v8f __attribute__((ext_vector_type(8)));

// exact pointee type the async-LDS builtin wants: GCC-style 4 x i32 vector
typedef int v4i __attribute__((vector_size(16)));
typedef __attribute__((address_space(1))) v4i gas_v4i;   // global (AS1)
typedef __attribute__((address_space(3))) v4i las_v4i;   // LDS (AS3)

__device__ __forceinline__ gas_v4i* to_global_v4(const void* p) {
    return (gas_v4i*)(unsigned long long)p;              // int->ptr reinterpret
}
__device__ __forceinline__ las_v4i* to_lds_v4(void* p) {
    // generic LDS address: low 32 bits are the LDS byte offset
    return (las_v4i*)(unsigned)(unsigned long long)p;
}

__device__ __forceinline__ float wave_max(float v) {
#pragma unroll
    for (int off = 16; off > 0; off >>= 1)
        v = fmaxf(v, __shfl_xor(v, off, 32));
    return v;
}
__device__ __forceinline__ float wave_sum(float v) {
#pragma unroll
    for (int off = 16; off > 0; off >>= 1)
        v += __shfl_xor(v, off, 32);
    return v;
}

// -------------------------------------------------------------------
// Kernel A: one row per block. Stage row into LDS (async on gfx1250),
// then max -> sum(exp) -> group sum -> per-row loss.
// -------------------------------------------------------------------
__global__ __launch_bounds__(NTHREADS)
void ls_row_loss_kernel(const float* __restrict__ logits,
                        const int*   __restrict__ target,
                        float*       __restrict__ row_loss) {
    __shared__ float sdata[N_CLASS];
    __shared__ float red[NWAVES];

    const int   row  = blockIdx.x;
    const int   tid  = threadIdx.x;
    const int   lane = tid & 31;
    const int   wave = tid >> 5;
    const float* rowp = logits + (size_t)row * N_CLASS;

    // ---- stage row into LDS ----
#if defined(__gfx1250__) && __has_builtin(__builtin_amdgcn_global_load_async_to_lds_b128)
    for (int idx = tid; idx < NVEC; idx += NTHREADS) {
        __builtin_amdgcn_global_load_async_to_lds_b128(
            to_global_v4(rowp + idx * 4), to_lds_v4(&sdata[idx * 4]),
            /*imm offset*/0, /*cpol*/0);
    }
#if __has_builtin(__builtin_amdgcn_s_wait_asynccnt)
    __builtin_amdgcn_s_wait_asynccnt(0);
#else
    asm volatile("s_wait_asynccnt 0" ::: "memory");
#endif
#else
    for (int idx = tid; idx < NVEC; idx += NTHREADS) {
        const float4 v = ((const float4*)rowp)[idx];
        ((float4*)sdata)[idx] = v;
    }
#endif
    __syncthreads();

    // ---- row max ----
    float m = -3.0e38f;
    for (int c = tid; c < N_CLASS; c += NTHREADS) m = fmaxf(m, sdata[c]);
    m = wave_max(m);
    if (lane == 0) red[wave] = m;
    __syncthreads();
    float rowmax = red[0];
#pragma unroll
    for (int w = 1; w < NWAVES; ++w) rowmax = fmaxf(rowmax, red[w]);
    __syncthreads();

    // ---- sum of exp(x - max) ----
    float s = 0.0f;
    for (int c = tid; c < N_CLASS; c += NTHREADS) s += expf(sdata[c] - rowmax);
    s = wave_sum(s);
    if (lane == 0) red[wave] = s;
    __syncthreads();
    float sumexp = red[0];
#pragma unroll
    for (int w = 1; w < NWAVES; ++w) sumexp += red[w];
    __syncthreads();

    // ---- group sum for the target's group ----
    const int tgt = target[row];
    const int lo  = (tgt / GROUP_SZ) * GROUP_SZ;
    float gs = 0.0f;
    for (int i = tid; i < GROUP_SZ; i += NTHREADS) gs += sdata[lo + i];
    gs = wave_sum(gs);
    if (lane == 0) red[wave] = gs;
    __syncthreads();

    if (tid == 0) {
        float gsum = red[0];
#pragma unroll
        for (int w = 1; w < NWAVES; ++w) gsum += red[w];
        const float xt  = sdata[tgt];
        const float lse = rowmax + logf(sumexp);
        row_loss[row] = lse - (CONF - SMOOTH_V) * xt - SMOOTH_V * gsum;
    }
}

// -------------------------------------------------------------------
// Kernel B: deterministic mean over 16384 per-row losses.
// Wave 0 folds 256 partials through 4 chained V_WMMA_F32_16X16X4_F32
// (B = all-ones => D[m,n] accumulates row sums; grand total recovered
// from the 8 accumulator VGPRs plus one lane-16 shuffle).
// -------------------------------------------------------------------
__global__ __launch_bounds__(256)
void ls_mean_kernel(const float* __restrict__ row_loss,
                    float*       __restrict__ out) {
    __shared__ float sp[256];
    const int tid = threadIdx.x;

    float s = 0.0f;
    for (int i = tid; i < BATCH; i += 256) s += row_loss[i];
    sp[tid] = s;
    __syncthreads();

    if (tid < 32) {
#if defined(__gfx1250__) && __has_builtin(__builtin_amdgcn_wmma_f32_16x16x4_f32)
        v8f c = {};
        v2f bones; bones.x = 1.0f; bones.y = 1.0f;   // B = ones (4x16)
#pragma unroll
        for (int i = 0; i < 4; ++i) {
            v2f a;                                    // 64 partials per WMMA
            a.x = sp[i * 64 + tid * 2];
            a.y = sp[i * 64 + tid * 2 + 1];
            c = __builtin_amdgcn_wmma_f32_16x16x4_f32(
                    false, a, false, bones, (short)0, c, false, false);
        }
        float t = 0.0f;
#pragma unroll
        for (int k = 0; k < 8; ++k) t += c[k];        // sum 8 accum VGPRs
        t += __shfl_xor(t, 16, 32);                   // rows 0-7 + rows 8-15
        if (tid == 0) out[0] = t * (1.0f / (float)BATCH);
#else
        float t = sp[tid];
#pragma unroll
        for (int w = 1; w < 8; ++w) t += sp[tid + w * 32];
        t = wave_sum(t);
        if (tid == 0) out[0] = t * (1.0f / (float)BATCH);
#endif
    }
}

// -------------------------------------------------------------------
extern "C" void kernel_launch(void* const* d_in, const int* in_sizes, int n_in,
                              void* d_out, int out_size, void* d_ws, size_t ws_size,
                              hipStream_t stream) {
    const float* logits = (const float*)d_in[0];
    const int*   target = (const int*)d_in[1];
    float*       rloss  = (float*)d_ws;              // 16384 floats of scratch

    ls_row_loss_kernel<<<BATCH, NTHREADS, 0, stream>>>(logits, target, rloss);
    ls_mean_kernel<<<1, 256, 0, stream>>>(rloss, (float*)d_out);
}